// KPConv_69148973466332
// MI455X (gfx1250) — compile-verified
//
#include <hip/hip_runtime.h>
#include <hip/hip_bf16.h>

typedef __attribute__((ext_vector_type(2))) float v2f;
typedef __attribute__((ext_vector_type(8))) float v8f;

#define FEAT 64   // IN == OUT == 64

// ---------------------------------------------------------------- zero S+deg
__global__ __launch_bounds__(256) void kp_zero(float4* __restrict__ p4, long n4,
                                               float* __restrict__ tail, int ntail) {
    long i = (long)blockIdx.x * blockDim.x + threadIdx.x;
    long stride = (long)gridDim.x * blockDim.x;
    float4 z = make_float4(0.f, 0.f, 0.f, 0.f);
    for (; i < n4; i += stride) p4[i] = z;
    if (blockIdx.x == 0 && threadIdx.x < (unsigned)ntail) tail[threadIdx.x] = 0.f;
}

// ------------------------------------------------- W[k,o,i] -> Wt[k,i,o]
__global__ __launch_bounds__(256) void kp_transpose_w(const float* __restrict__ W,
                                                      float* __restrict__ Wt, int total) {
    int t = blockIdx.x * 256 + threadIdx.x;
    if (t >= total) return;
    int k = t >> 12;        // /4096
    int r = t & 4095;
    int o = r >> 6;
    int i = r & 63;
    Wt[(k << 12) + (i << 6) + o] = W[t];
}

// ---------------------------------------------------------- per-edge scatter
// one wave32 per edge
__global__ __launch_bounds__(256) void kp_edge(const float* __restrict__ x,
                                               const float* __restrict__ pos,
                                               const float* __restrict__ kpts,
                                               const int* __restrict__ ei,
                                               float* __restrict__ S,
                                               float* __restrict__ deg,
                                               int E, int Kk) {
    const int lane = threadIdx.x & 31;
    const int e = blockIdx.x * (blockDim.x >> 5) + (threadIdx.x >> 5);
    if (e >= E) return;

    const int src = ei[e];
    const int dst = ei[E + e];

    const float rx = pos[src * 3 + 0] - pos[dst * 3 + 0];
    const float ry = pos[src * 3 + 1] - pos[dst * 3 + 1];
    const float rz = pos[src * 3 + 2] - pos[dst * 3 + 2];

    // lanes 0..K-1 own one kernel point each
    float d = 3.0e38f;
    if (lane < Kk) {
        const float dx = rx - kpts[lane * 3 + 0];
        const float dy = ry - kpts[lane * 3 + 1];
        const float dz = rz - kpts[lane * 3 + 2];
        d = dx * dx + dy * dy + dz * dz;
    }
    int idx = lane;
    // butterfly argmin over the wave; first-index tie-break (matches jnp.argmin)
    #pragma unroll
    for (int off = 16; off > 0; off >>= 1) {
        const float od = __shfl_xor(d, off, 32);
        const int   oi = __shfl_xor(idx, off, 32);
        if (od < d || (od == d && oi < idx)) { d = od; idx = oi; }
    }
    float w = fmaxf(1.0f - sqrtf(d) * 5.0f, 0.0f);   // 1 - sqrt(d)/0.2

    // 64 features: 2 per lane, contiguous 256B region -> coalesced atomics
    const float2 f = *(const float2*)(x + (size_t)src * FEAT + lane * 2);
    float* sp = S + ((size_t)dst * Kk + idx) * FEAT + lane * 2;
    unsafeAtomicAdd(sp + 0, w * f.x);
    unsafeAtomicAdd(sp + 1, w * f.y);

    if (lane == 0) unsafeAtomicAdd(deg + dst, 1.0f);
}

// ------------------------------------------------------------- WMMA f32 GEMM
// out[n,o] = (1/max(deg,1)) * sum_{c=0..K*64} S[n,c] * Wt[c,o]
// per wave: 16 rows x 64 cols, 4 accumulators (4 col-tiles of 16)
__global__ __launch_bounds__(256) void kp_gemm(const float* __restrict__ S,
                                               const float* __restrict__ Wt,
                                               const float* __restrict__ deg,
                                               float* __restrict__ out,
                                               int N, int contr /* = K*64 */) {
    const int lane = threadIdx.x & 31;
    const int wave = threadIdx.x >> 5;
    const int n0 = (blockIdx.x * 8 + wave) * 16;
    if (n0 >= N) return;

    const int mrow  = lane & 15;            // A-row / C-col index
    const int khalf = (lane >> 4) << 1;     // 0 or 2: which K-pair this half-wave holds

    const int  row      = n0 + mrow;
    const bool rowvalid = row < N;
    const float* arow = S + (size_t)(rowvalid ? row : (N - 1)) * contr;

    v8f c0 = {}, c1 = {}, c2 = {}, c3 = {};

    for (int c = 0; c < contr; c += 4) {
        // A fragment (16x4 f32): lane holds A[mrow][khalf], A[mrow][khalf+1]
        v2f a;
        {
            const float ax = arow[c + khalf + 0];
            const float ay = arow[c + khalf + 1];
            a.x = rowvalid ? ax : 0.0f;
            a.y = rowvalid ? ay : 0.0f;
        }
        // B fragments (4x16 f32), mirrored layout, from transposed weights
        const float* b0p = Wt + (size_t)(c + khalf) * FEAT + mrow;
        const float* b1p = b0p + FEAT;      // next K row
        v2f b0, b1, b2, b3;
        b0.x = b0p[0];  b0.y = b1p[0];
        b1.x = b0p[16]; b1.y = b1p[16];
        b2.x = b0p[32]; b2.y = b1p[32];
        b3.x = b0p[48]; b3.y = b1p[48];

        c0 = __builtin_amdgcn_wmma_f32_16x16x4_f32(false, a, false, b0, (short)0, c0, false, false);
        c1 = __builtin_amdgcn_wmma_f32_16x16x4_f32(false, a, false, b1, (short)0, c1, false, false);
        c2 = __builtin_amdgcn_wmma_f32_16x16x4_f32(false, a, false, b2, (short)0, c2, false, false);
        c3 = __builtin_amdgcn_wmma_f32_16x16x4_f32(false, a, false, b3, (short)0, c3, false, false);
    }

    // C/D layout: lane col = lane&15; VGPR v holds M = v + 8*(lane>=16)
    const int half = (lane >> 4) << 3;
    #pragma unroll
    for (int v = 0; v < 8; ++v) {
        const int m = n0 + v + half;
        if (m < N) {
            const float sc = 1.0f / fmaxf(deg[m], 1.0f);
            float* orow = out + (size_t)m * FEAT + mrow;
            orow[0]  = c0[v] * sc;
            orow[16] = c1[v] * sc;
            orow[32] = c2[v] * sc;
            orow[48] = c3[v] * sc;
        }
    }
}

// ---------------------------------------------------------------------------
extern "C" void kernel_launch(void* const* d_in, const int* in_sizes, int n_in,
                              void* d_out, int out_size, void* d_ws, size_t ws_size,
                              hipStream_t stream) {
    const float* x    = (const float*)d_in[0];
    const float* pos  = (const float*)d_in[1];
    const float* kpts = (const float*)d_in[2];
    const float* Kw   = (const float*)d_in[3];
    const int*   ei   = (const int*)d_in[4];
    float* out = (float*)d_out;

    const int N  = in_sizes[0] / FEAT;   // 50000
    const int Kk = in_sizes[2] / 3;      // 15
    const int E  = in_sizes[4] / 2;      // 800000
    const int contr = Kk * FEAT;         // 960

    // workspace layout: S [N*K*64] f32 | deg [N] f32 | Wt [K*64*64] f32
    float* S   = (float*)d_ws;
    const size_t s_elems = (size_t)N * Kk * FEAT;
    float* deg = S + s_elems;
    float* Wt  = deg + N;

    // 1) zero S + deg
    const long  ztotal = (long)s_elems + N;
    const long  n4     = ztotal >> 2;
    const int   ntail  = (int)(ztotal & 3);
    kp_zero<<<4096, 256, 0, stream>>>((float4*)S, n4, S + (n4 << 2), ntail);

    // 2) transpose weights for contiguous WMMA B loads
    const int wtotal = Kk * FEAT * FEAT;
    kp_transpose_w<<<(wtotal + 255) / 256, 256, 0, stream>>>(Kw, Wt, wtotal);

    // 3) per-edge geometry + weighted scatter (one wave per edge)
    kp_edge<<<(E + 7) / 8, 256, 0, stream>>>(x, pos, kpts, ei, S, deg, E, Kk);

    // 4) bucketed GEMM via v_wmma_f32_16x16x4_f32 + degree normalization
    kp_gemm<<<(N + 127) / 128, 256, 0, stream>>>(S, Wt, deg, out, N, contr);
}